// CRM_57861799411851
// MI455X (gfx1250) — compile-verified
//
#include <hip/hip_runtime.h>
#include <hip/hip_bf16.h>
#include <stdint.h>

#define HH 1536
#define WW 2048
#define SS 2000
#define SIDE_R 20
#define THR 0.2f

typedef unsigned int uint32x4 __attribute__((ext_vector_type(4)));
typedef int int32x8 __attribute__((ext_vector_type(8)));
typedef int int32x4 __attribute__((ext_vector_type(4)));
typedef float v2f __attribute__((ext_vector_type(2)));
typedef float v8f __attribute__((ext_vector_type(8)));

#if __has_builtin(__builtin_amdgcn_tensor_load_to_lds)
#define HAVE_TDM 1
#else
#define HAVE_TDM 0
#endif

#if __has_builtin(__builtin_amdgcn_wmma_f32_16x16x4_f32)
#define HAVE_WMMA 1
#else
#define HAVE_WMMA 0
#endif

struct F3 { float x, y, z; };
__device__ __forceinline__ F3 mkf3(float a, float b, float c) { F3 r; r.x = a; r.y = b; r.z = c; return r; }
__device__ __forceinline__ F3 f3sub(F3 a, F3 b) { return mkf3(a.x - b.x, a.y - b.y, a.z - b.z); }
__device__ __forceinline__ F3 f3cross(F3 a, F3 b) {
    return mkf3(a.y * b.z - a.z * b.y, a.z * b.x - a.x * b.z, a.x * b.y - a.y * b.x);
}
__device__ __forceinline__ F3 f3normeps(F3 a) {  // v / (||v|| + 1e-6), matches _normalize
    float n = sqrtf(a.x * a.x + a.y * a.y + a.z * a.z) + 1e-6f;
    return mkf3(a.x / n, a.y / n, a.z / n);
}
__device__ __forceinline__ F3 camPoint(const float* k, float d, int x, int y) {
    float u = (float)x, v = (float)y;
    return mkf3((k[0] * u + k[1] * v + k[2]) * d,
                (k[3] * u + k[4] * v + k[5]) * d,
                (k[6] * u + k[7] * v + k[8]) * d);
}

__global__ void zeroKernel(float* p, int n) {
    int i = blockIdx.x * blockDim.x + threadIdx.x;
    if (i < n) p[i] = 0.0f;
}

// Kernel A: fused backproject + surface normals + side-region flags.
// TDM prefetches the block's depth halo tile into LDS (async-tensor path);
// compute reads via the (now warm) cache path for guaranteed correctness.
__global__ void normalsRegionKernel(const float* __restrict__ depth,
                                    const int* __restrict__ mask,
                                    const float* __restrict__ Kinv,
                                    float* __restrict__ nxp, float* __restrict__ nyp,
                                    float* __restrict__ nzp,
                                    unsigned char* __restrict__ flags) {
#if HAVE_TDM
    __shared__ float tdm_buf[34 * 10];
    if (threadIdx.y == 0) {  // one TDM issue per block (wave 0 only)
        int bx0 = blockIdx.x * 32, by0 = blockIdx.y * 8;
        int gx0 = bx0 > 0 ? bx0 - 1 : 0;
        int gy0 = by0 > 0 ? by0 - 1 : 0;
        int gx1 = (bx0 + 33 < WW) ? bx0 + 33 : WW;
        int gy1 = (by0 + 9 < HH) ? by0 + 9 : HH;
        unsigned tw = (unsigned)(gx1 - gx0);
        unsigned th2 = (unsigned)(gy1 - gy0);
        unsigned long long gaddr =
            (unsigned long long)(uintptr_t)(const void*)(depth + (size_t)gy0 * WW + gx0);
        unsigned ldsoff = (unsigned)(uintptr_t)(void*)&tdm_buf[0];  // flat LDS addr low 32b = offset
        // D# group0: count=1 | lds_addr | global_addr[56:0] | type=2
        uint32x4 g0;
        g0.x = 1u;
        g0.y = ldsoff;
        g0.z = (unsigned)(gaddr & 0xFFFFFFFFull);
        g0.w = ((unsigned)((gaddr >> 32) & 0x01FFFFFFull)) | (2u << 30);
        // D# group1: data_size=4B; tensor/tile dims = clamped extents (no OOB); stride = W.
        int32x8 g1;
        g1.s0 = (int)(2u << 16);
        g1.s1 = (int)((tw & 0xFFFFu) << 16);
        g1.s2 = (int)(((tw >> 16) & 0xFFFFu) | ((th2 & 0xFFFFu) << 16));
        g1.s3 = (int)(((th2 >> 16) & 0xFFFFu) | ((tw & 0xFFFFu) << 16));
        g1.s4 = (int)(th2 & 0xFFFFu);
        g1.s5 = (int)WW;
        g1.s6 = 0;
        g1.s7 = 0;
        int32x4 gz = {0, 0, 0, 0};  // 2D tensor: groups 2/3 unused
#if defined(__clang_major__) && (__clang_major__ >= 23)
        int32x8 gzz = {0, 0, 0, 0, 0, 0, 0, 0};
        __builtin_amdgcn_tensor_load_to_lds(g0, g1, gz, gz, gzz, 0);
#else
        __builtin_amdgcn_tensor_load_to_lds(g0, g1, gz, gz, 0);
#endif
#if __has_builtin(__builtin_amdgcn_s_wait_tensorcnt)
        __builtin_amdgcn_s_wait_tensorcnt(0);
#endif
    }
#endif
    int x = blockIdx.x * 32 + threadIdx.x;
    int y = blockIdx.y * 8 + threadIdx.y;
    if (x >= WW || y >= HH) return;
    float kk[9];
#pragma unroll
    for (int i = 0; i < 9; ++i) kk[i] = Kinv[i];
    size_t idx = (size_t)y * WW + x;

    float nxv = 0.0f, nyv = 0.0f, nzv = 0.0f;
    if (x > 0 && x < WW - 1 && y > 0 && y < HH - 1) {
        const float* dr = depth;
        F3 ctr = camPoint(kk, dr[idx], x, y);
        F3 xm = camPoint(kk, dr[idx - 1], x - 1, y);
        F3 xp = camPoint(kk, dr[idx + 1], x + 1, y);
        F3 ym = camPoint(kk, dr[idx - WW], x, y - 1);
        F3 yp = camPoint(kk, dr[idx + WW], x, y + 1);
        F3 mm = camPoint(kk, dr[idx - WW - 1], x - 1, y - 1);
        F3 mp = camPoint(kk, dr[idx + WW - 1], x - 1, y + 1);
        F3 pm = camPoint(kk, dr[idx - WW + 1], x + 1, y - 1);
        F3 pp = camPoint(kk, dr[idx + WW + 1], x + 1, y + 1);
        F3 n0 = f3normeps(f3cross(f3sub(xm, ctr), f3sub(ym, ctr)));
        F3 n1 = f3normeps(f3cross(f3sub(xp, ctr), f3sub(yp, ctr)));
        F3 n2 = f3normeps(f3cross(f3sub(mm, ctr), f3sub(mp, ctr)));
        F3 n3 = f3normeps(f3cross(f3sub(pm, ctr), f3sub(pp, ctr)));
        F3 s = mkf3((n0.x + n1.x + n2.x + n3.x) * 0.25f,
                    (n0.y + n1.y + n2.y + n3.y) * 0.25f,
                    (n0.z + n1.z + n2.z + n3.z) * 0.25f);
        F3 nf = f3normeps(s);
        nxv = nf.x; nyv = nf.y; nzv = nf.z;
    }
    nxp[idx] = nxv; nyp[idx] = nyv; nzp[idx] = nzv;

    // side regions: dilated left/right mask edges AND ~mask
    unsigned char f = 0;
    if (mask[idx] == 0) {
        int xa = x - SIDE_R; if (xa < 0) xa = 0;
        int xb = x + SIDE_R; if (xb > WW - 1) xb = WW - 1;
        bool L = false, R = false;
        int mc = (mask[(size_t)y * WW + xa] != 0);
        for (int xi = xa; xi <= xb; ++xi) {
            int mn = (xi + 1 < WW) ? (mask[(size_t)y * WW + xi + 1] != 0) : 0;
            if (!mc && mn) L = true;   // grad_x == +1 -> left edge
            if (mc && !mn) R = true;   // grad_x == -1 -> right edge
            mc = mn;
            if (L && R) break;
        }
        f = (unsigned char)((L ? 1u : 0u) | (R ? 2u : 0u));
    }
    flags[idx] = f;
}

// Kernel A2: deterministic per-row region-normal partial sums (fixed-order tree reduce).
// Output layout: rowAcc8[y*8 + {Lx,Ly,Lz,Lcnt,Rx,Ry,Rz,Rcnt}] (feeds the WMMA matvec).
__global__ void rowAccKernel(const float* __restrict__ nxp, const float* __restrict__ nyp,
                             const float* __restrict__ nzp, const unsigned char* __restrict__ flags,
                             float* __restrict__ rowAcc8) {
    int y = blockIdx.x;
    float a[8] = {0, 0, 0, 0, 0, 0, 0, 0};
    for (int x = threadIdx.x; x < WW; x += blockDim.x) {
        size_t idx = (size_t)y * WW + x;
        unsigned char f = flags[idx];
        if (f & 3u) {
            float vx = nxp[idx], vy = nyp[idx], vz = nzp[idx];
            if (f & 1u) { a[0] += vx; a[1] += vy; a[2] += vz; a[3] += 1.0f; }
            if (f & 2u) { a[4] += vx; a[5] += vy; a[6] += vz; a[7] += 1.0f; }
        }
    }
    __shared__ float red[256 * 8];
    int t = threadIdx.x;
#pragma unroll
    for (int k = 0; k < 8; ++k) red[t * 8 + k] = a[k];
    __syncthreads();
    for (int s2 = 128; s2 > 0; s2 >>= 1) {
        if (t < s2)
#pragma unroll
            for (int k = 0; k < 8; ++k) red[t * 8 + k] += red[(t + s2) * 8 + k];
        __syncthreads();
    }
    if (t == 0) {
#pragma unroll
        for (int k = 0; k < 8; ++k) rowAcc8[y * 8 + k] = red[k];
    }
}

// Kernel B: sums(8) = ones(1 x HH) . rowAcc8(HH x 8) as a WMMA f32 16x16x4 matvec chain,
// then finalize normalized mean normals (+ cos denominator). Single wave (32 threads),
// EXEC all-ones at the WMMA, fixed iteration order -> deterministic.
__global__ void meanKernel(const float* __restrict__ rowAcc8, float* __restrict__ meanbuf) {
    __shared__ float tmp[16];
    int lane = threadIdx.x;
#if HAVE_WMMA
    // A (16x4 f32): row M=0 all-ones, other rows zero.
    // ISA layout: lanes 0-15 = M, VGPR0 = K={0,2}, VGPR1 = K={1,3} -> lane 0 & lane 16 hold M=0.
    float av = (lane == 0 || lane == 16) ? 1.0f : 0.0f;
    v2f a; a.x = av; a.y = av;
    // B (4x16 f32): columns 0..7 = the 8 accumulators, rows = 4 consecutive image rows.
    // Any K-labeling permutation is harmless: A row0 is all-ones, so D[0][n] = sum_k B[k][n].
    int q = lane & 15;
    int hi = lane >> 4;
    int col = q & 7;
    float colmask = (q < 8) ? 1.0f : 0.0f;
    v8f c = {0, 0, 0, 0, 0, 0, 0, 0};
    for (int y0 = 0; y0 < HH; y0 += 4) {
        v2f b;
        b.x = rowAcc8[(y0 + 2 * hi + 0) * 8 + col] * colmask;
        b.y = rowAcc8[(y0 + 2 * hi + 1) * 8 + col] * colmask;
        c = __builtin_amdgcn_wmma_f32_16x16x4_f32(false, a, false, b, (short)0, c,
                                                  false, false);
    }
    // D layout: VGPR0, lanes 0-15 = row M=0, N=lane -> the 8 column sums live in lanes 0..7.
    if (lane < 16) tmp[lane] = c[0];
    __syncthreads();
#else
    if (lane == 0) {
        for (int k = 0; k < 8; ++k) {
            float s = 0.0f;
            for (int y = 0; y < HH; ++y) s += rowAcc8[y * 8 + k];
            tmp[k] = s;
        }
    }
    __syncthreads();
#endif
    if (lane == 0) {
        for (int side = 0; side < 2; ++side) {
            float sx = tmp[side * 4 + 0], sy = tmp[side * 4 + 1];
            float sz = tmp[side * 4 + 2], cn = tmp[side * 4 + 3];
            float cc = fmaxf(cn, 1.0f);
            float mx = sx / cc, my = sy / cc, mz = sz / cc;
            float n = sqrtf(mx * mx + my * my + mz * mz);
            float n12 = fmaxf(n, 1e-12f);
            mx /= n12; my /= n12; mz /= n12;
            float denom = fmaxf(sqrtf(mx * mx + my * my + mz * mz), 1e-8f);
            meanbuf[side * 4 + 0] = mx; meanbuf[side * 4 + 1] = my;
            meanbuf[side * 4 + 2] = mz; meanbuf[side * 4 + 3] = denom;
        }
    }
}

// Kernel C: inlier flags (cos > THR) + per-row inlier / mask counts (int-exact reductions).
__global__ void inlierKernel(const float* __restrict__ nxp, const float* __restrict__ nyp,
                             const float* __restrict__ nzp, unsigned char* __restrict__ flags,
                             const int* __restrict__ mask, const float* __restrict__ meanbuf,
                             int* __restrict__ rowCntL, int* __restrict__ rowCntR,
                             int* __restrict__ rowMaskCnt) {
    int y = blockIdx.x;
    float mLx = meanbuf[0], mLy = meanbuf[1], mLz = meanbuf[2], dL = meanbuf[3];
    float mRx = meanbuf[4], mRy = meanbuf[5], mRz = meanbuf[6], dR = meanbuf[7];
    int cl = 0, cr = 0, cm = 0;
    for (int x = threadIdx.x; x < WW; x += blockDim.x) {
        size_t idx = (size_t)y * WW + x;
        cm += (mask[idx] != 0) ? 1 : 0;
        unsigned char f = flags[idx];
        if (f & 3u) {
            float vx = nxp[idx], vy = nyp[idx], vz = nzp[idx];
            float nn = fmaxf(sqrtf(vx * vx + vy * vy + vz * vz), 1e-8f);
            if (f & 1u) {
                float cosv = (vx * mLx + vy * mLy + vz * mLz) / (nn * dL);
                if (cosv > THR) { f |= 4u; ++cl; }
            }
            if (f & 2u) {
                float cosv = (vx * mRx + vy * mRy + vz * mRz) / (nn * dR);
                if (cosv > THR) { f |= 8u; ++cr; }
            }
            flags[idx] = f;
        }
    }
    __shared__ int red[256 * 3];
    int t = threadIdx.x;
    red[t * 3 + 0] = cl; red[t * 3 + 1] = cr; red[t * 3 + 2] = cm;
    __syncthreads();
    for (int s2 = 128; s2 > 0; s2 >>= 1) {
        if (t < s2) {
            red[t * 3 + 0] += red[(t + s2) * 3 + 0];
            red[t * 3 + 1] += red[(t + s2) * 3 + 1];
            red[t * 3 + 2] += red[(t + s2) * 3 + 2];
        }
        __syncthreads();
    }
    if (t == 0) { rowCntL[y] = red[0]; rowCntR[y] = red[1]; rowMaskCnt[y] = red[2]; }
}

// Kernel D: serial exclusive scan over 1536 rows (trivial work, deterministic).
__global__ void scanKernel(const int* __restrict__ rowCntL, const int* __restrict__ rowCntR,
                           int* __restrict__ rowStartL, int* __restrict__ rowStartR) {
    if (threadIdx.x != 0 || blockIdx.x != 0) return;
    int s = 0;
    for (int y = 0; y < HH; ++y) { rowStartL[y] = s; s += rowCntL[y]; }
    rowStartL[HH] = s;
    s = 0;
    for (int y = 0; y < HH; ++y) { rowStartR[y] = s; s += rowCntR[y]; }
    rowStartR[HH] = s;
}

// Kernel E: rank-select the S evenly spaced inliers (== sort+gather of the reference)
// and accumulate per-row segment sums. Atomic float adds of small integers are exact.
__global__ void sampleKernel(const unsigned char* __restrict__ flags,
                             const int* __restrict__ rowStartL, const int* __restrict__ rowStartR,
                             float* __restrict__ ls, float* __restrict__ lc,
                             float* __restrict__ rs, float* __restrict__ rc) {
    int i = blockIdx.x * blockDim.x + threadIdx.x;
    if (i >= 2 * SS) return;
    int side = (i >= SS) ? 1 : 0;
    int j = side ? i - SS : i;
    const int* rowStart = side ? rowStartR : rowStartL;
    unsigned char bit = side ? 8u : 4u;
    int N = rowStart[HH];
    int y, x;
    if (N <= 0) {  // srt[pos] == H*W, clipped to H*W-1
        y = HH - 1; x = WW - 1;
    } else {
        int pos;
        if (N >= SS) pos = (int)floorf((float)j * (float)(N - 1) / (float)(SS - 1));
        else pos = (j < N - 1) ? j : (N - 1);
        int a = 0, b = HH;  // rowStart[a] <= pos < rowStart[b]
        while (b - a > 1) { int m = (a + b) >> 1; if (rowStart[m] <= pos) a = m; else b = m; }
        y = a;
        int need = pos - rowStart[y];
        x = 0; int cnt = 0;
        const unsigned char* rowf = flags + (size_t)y * WW;
        for (int xi = 0; xi < WW; ++xi) {
            if (rowf[xi] & bit) { if (cnt == need) { x = xi; break; } ++cnt; }
        }
    }
    atomicAdd(side ? &rs[y] : &ls[y], (float)x);
    atomicAdd(side ? &rc[y] : &lc[y], 1.0f);
}

// Kernel F: per-row interpolation endpoints + validity.
__global__ void rowParamKernel(const float* __restrict__ depth,
                               const int* __restrict__ rowStartL, const int* __restrict__ rowStartR,
                               const float* __restrict__ ls, const float* __restrict__ lc,
                               const float* __restrict__ rs, const float* __restrict__ rc,
                               const int* __restrict__ rowMaskCnt,
                               float* __restrict__ rlx, float* __restrict__ rrx,
                               float* __restrict__ rld, float* __restrict__ rrd,
                               int* __restrict__ rvalid) {
    int y = blockIdx.x * blockDim.x + threadIdx.x;
    if (y >= HH) return;
    float lcv = lc[y], rcv = rc[y];
    int Nl = rowStartL[HH], Nr = rowStartR[HH];
    float lmean = ls[y] / fmaxf(lcv, 1.0f);
    float rmean = rs[y] / fmaxf(rcv, 1.0f);
    int lx = (int)rintf(lmean);  // jnp.round = half-to-even
    int rx = (int)rintf(rmean);
    lx = lx < 0 ? 0 : (lx > WW - 1 ? WW - 1 : lx);
    rx = rx < 0 ? 0 : (rx > WW - 1 ? WW - 1 : rx);
    float ld = depth[(size_t)y * WW + lx];
    float rd = depth[(size_t)y * WW + rx];
    bool valid = (lcv > 0.f) && (rcv > 0.f) && (Nl > 0) && (Nr > 0) &&
                 !(ld != ld) && !(rd != rd) && (rowMaskCnt[y] > 0);
    rlx[y] = (float)lx; rrx[y] = (float)rx; rld[y] = ld; rrd[y] = rd;
    rvalid[y] = valid ? 1 : 0;
}

// Kernel G: streaming interpolation into the mask.
__global__ void outputKernel(const float* __restrict__ depth, const int* __restrict__ mask,
                             const float* __restrict__ rlx, const float* __restrict__ rrx,
                             const float* __restrict__ rld, const float* __restrict__ rrd,
                             const int* __restrict__ rvalid, float* __restrict__ out) {
    int idx = blockIdx.x * blockDim.x + threadIdx.x;
    if (idx >= HH * WW) return;
    if (idx + 4096 < HH * WW) __builtin_prefetch(&depth[idx + 4096], 0, 0);  // global_prefetch_b8
    int y = idx / WW;
    int x = idx - y * WW;
    float d = depth[idx];
    float o = d;
    if (mask[idx] != 0 && rvalid[y] != 0) {
        float lxf = rlx[y], rxf = rrx[y];
        float ratio = ((float)x - lxf) / (rxf - lxf + 1e-6f);
        ratio = fminf(fmaxf(ratio, 0.0f), 1.0f);
        o = (1.0f - ratio) * rld[y] + ratio * rrd[y];
    }
    out[idx] = o;
}

extern "C" void kernel_launch(void* const* d_in, const int* in_sizes, int n_in,
                              void* d_out, int out_size, void* d_ws, size_t ws_size,
                              hipStream_t stream) {
    const float* depth = (const float*)d_in[0];
    const int* mask = (const int*)d_in[1];
    const float* Kinv = (const float*)d_in[2];
    float* out = (float*)d_out;
    (void)in_sizes; (void)n_in; (void)out_size; (void)ws_size;

    const size_t HWsz = (size_t)HH * WW;
    char* w = (char*)d_ws;
    float* nxp = (float*)w;                   w += HWsz * sizeof(float);
    float* nyp = (float*)w;                   w += HWsz * sizeof(float);
    float* nzp = (float*)w;                   w += HWsz * sizeof(float);
    unsigned char* flags = (unsigned char*)w; w += HWsz;
    w = (char*)(((uintptr_t)w + 255) & ~(uintptr_t)255);
    float* rowAcc8 = (float*)w;   w += (size_t)HH * 8 * sizeof(float);
    float* meanbuf = (float*)w;   w += 16 * sizeof(float);
    int* rowCntL = (int*)w;       w += (size_t)HH * sizeof(int);
    int* rowCntR = (int*)w;       w += (size_t)HH * sizeof(int);
    int* rowMaskCnt = (int*)w;    w += (size_t)HH * sizeof(int);
    int* rowStartL = (int*)w;     w += (size_t)(HH + 1) * sizeof(int);
    int* rowStartR = (int*)w;     w += (size_t)(HH + 1) * sizeof(int);
    w = (char*)(((uintptr_t)w + 255) & ~(uintptr_t)255);
    float* segsum = (float*)w;    w += (size_t)4 * HH * sizeof(float);  // ls,lc,rs,rc contiguous
    float* ls = segsum, *lc = segsum + HH, *rs = segsum + 2 * HH, *rc = segsum + 3 * HH;
    float* rlx = (float*)w;       w += (size_t)HH * sizeof(float);
    float* rrx = (float*)w;       w += (size_t)HH * sizeof(float);
    float* rld = (float*)w;       w += (size_t)HH * sizeof(float);
    float* rrd = (float*)w;       w += (size_t)HH * sizeof(float);
    int* rvalid = (int*)w;        w += (size_t)HH * sizeof(int);

    zeroKernel<<<(4 * HH + 255) / 256, 256, 0, stream>>>(segsum, 4 * HH);
    normalsRegionKernel<<<dim3(WW / 32, HH / 8), dim3(32, 8), 0, stream>>>(
        depth, mask, Kinv, nxp, nyp, nzp, flags);
    rowAccKernel<<<HH, 256, 0, stream>>>(nxp, nyp, nzp, flags, rowAcc8);
    meanKernel<<<1, 32, 0, stream>>>(rowAcc8, meanbuf);
    inlierKernel<<<HH, 256, 0, stream>>>(nxp, nyp, nzp, flags, mask, meanbuf,
                                         rowCntL, rowCntR, rowMaskCnt);
    scanKernel<<<1, 1, 0, stream>>>(rowCntL, rowCntR, rowStartL, rowStartR);
    sampleKernel<<<(2 * SS + 255) / 256, 256, 0, stream>>>(flags, rowStartL, rowStartR,
                                                           ls, lc, rs, rc);
    rowParamKernel<<<(HH + 255) / 256, 256, 0, stream>>>(depth, rowStartL, rowStartR,
                                                         ls, lc, rs, rc, rowMaskCnt,
                                                         rlx, rrx, rld, rrd, rvalid);
    outputKernel<<<((int)HWsz + 255) / 256, 256, 0, stream>>>(depth, mask, rlx, rrx,
                                                              rld, rrd, rvalid, out);
}